// Encoder_26053271618045
// MI455X (gfx1250) — compile-verified
//
#include <hip/hip_runtime.h>

// Problem constants (fixed by the reference)
#define N1c 40960
#define N2c 4096
#define Hc  256
#define DINc 128

typedef __attribute__((ext_vector_type(2))) float v2f;
typedef __attribute__((ext_vector_type(8))) float v8f;

__global__ void zero_kernel(float* __restrict__ p, long n) {
  long i = (long)blockIdx.x * blockDim.x + threadIdx.x;
  long stride = (long)gridDim.x * blockDim.x;
  for (; i < n; i += stride) p[i] = 0.0f;
}

// One wave32 per edge. Lanes stride the K-dim features: coalesced row load of
// x[src], coalesced global_atomic_add_f32 into agg[dst]. Lane 0 bumps count.
__global__ void scatter_kernel(const float* __restrict__ xsrc,
                               const int* __restrict__ src,
                               const int* __restrict__ dst,
                               float* __restrict__ agg,
                               float* __restrict__ cnt,
                               int E, int K) {
  long t = (long)blockIdx.x * blockDim.x + threadIdx.x;
  int edge = (int)(t >> 5);
  int lane = (int)(t & 31);
  if (edge >= E) return;
  int s = src[edge];
  int d = dst[edge];
  const float* xr = xsrc + (long)s * K;
  float* ar = agg + (long)d * K;
  for (int f = lane; f < K; f += 32) {
    atomicAdd(&ar[f], xr[f]);
  }
  if (lane == 0) atomicAdd(&cnt[d], 1.0f);
}

// out[M, 256] = act( (agg/max(cnt,1)) @ Wl^T + bias + Xd @ Wr^T )
// One wave per 16x64 output strip (4 n-tiles). WMMA f32 16x16x4, K stepped by 4.
// A fragment (mean row + root row) is loaded ONCE per K-chunk and reused across
// the 4 n-tiles -> A-array L2 traffic cut 4x vs per-tile loading; weights are
// tiny and L2-resident.
// A 16x4 f32 layout: lanes 0-15 -> M=lane, K={k0,k0+1}; lanes 16-31 -> K={k0+2,k0+3}
// B 4x16 f32 layout: lanes 0-15 -> N=lane, K={k0,k0+1}; lanes 16-31 -> K={k0+2,k0+3}
// D 16x16 f32: vgpr r, lanes 0-15 -> (M=r, N=lane); lanes 16-31 -> (M=8+r, N=lane-16)
__global__ __launch_bounds__(256) void sage_gemm_wmma(
    const float* __restrict__ agg, const float* __restrict__ cnt,
    const float* __restrict__ xdst,
    const float* __restrict__ Wl, const float* __restrict__ Wr,
    const float* __restrict__ bias,
    float* __restrict__ out,
    int K, int relu) {
  const int NGRP = Hc / 64;  // 4 strip groups of 4 n-tiles each
  int wave  = (int)((blockIdx.x * blockDim.x + threadIdx.x) >> 5);
  int lane  = threadIdx.x & 31;
  int mtile = wave / NGRP;
  int ngrp  = wave % NGRP;   // covers n-tiles 4*ngrp .. 4*ngrp+3
  int l15   = lane & 15;
  int khalf = lane >> 4;

  int row = mtile * 16 + l15;  // A-matrix row this lane loads
  float invc = 1.0f / fmaxf(cnt[row], 1.0f);

  const float* aL = agg  + (long)row * K + 2 * khalf;  // mean path A
  const float* aR = xdst + (long)row * K + 2 * khalf;  // root path A

  const float* bL[4];
  const float* bR[4];
  #pragma unroll
  for (int t = 0; t < 4; ++t) {
    int ncol = ngrp * 64 + t * 16 + l15;  // B column this lane loads
    bL[t] = Wl + (long)ncol * K + 2 * khalf;
    bR[t] = Wr + (long)ncol * K + 2 * khalf;
  }

  v8f acc[4] = {};
  for (int k0 = 0; k0 < K; k0 += 4) {
    float2 va = *(const float2*)(aL + k0);
    float2 vr = *(const float2*)(aR + k0);
    v2f aMean; aMean[0] = va.x * invc; aMean[1] = va.y * invc;
    v2f aRoot; aRoot[0] = vr.x;        aRoot[1] = vr.y;
    #pragma unroll
    for (int t = 0; t < 4; ++t) {
      float2 wb = *(const float2*)(bL[t] + k0);
      float2 wc = *(const float2*)(bR[t] + k0);
      v2f b1v; b1v[0] = wb.x; b1v[1] = wb.y;
      v2f b2v; b2v[0] = wc.x; b2v[1] = wc.y;
      acc[t] = __builtin_amdgcn_wmma_f32_16x16x4_f32(false, aMean, false, b1v,
                                                     (short)0, acc[t], false, false);
      acc[t] = __builtin_amdgcn_wmma_f32_16x16x4_f32(false, aRoot, false, b2v,
                                                     (short)0, acc[t], false, false);
    }
  }

  #pragma unroll
  for (int t = 0; t < 4; ++t) {
    int ncol = ngrp * 64 + t * 16 + l15;
    float bv = bias[ncol];
    #pragma unroll
    for (int r = 0; r < 8; ++r) {
      float v = acc[t][r] + bv;
      if (relu) v = fmaxf(v, 0.0f);
      int orow = mtile * 16 + r + 8 * khalf;
      out[(long)orow * Hc + ncol] = v;
    }
  }
}

extern "C" void kernel_launch(void* const* d_in, const int* in_sizes, int n_in,
                              void* d_out, int out_size, void* d_ws, size_t ws_size,
                              hipStream_t stream) {
  (void)n_in; (void)out_size; (void)ws_size;

  const float* x    = (const float*)d_in[0];
  const int*   src1 = (const int*)d_in[1];
  const int*   dst1 = (const int*)d_in[2];
  const int*   src2 = (const int*)d_in[3];
  const int*   dst2 = (const int*)d_in[4];
  // d_in[5], d_in[6] are n1, n2 scalars (device side); compile-time constants used.
  const float* W1l  = (const float*)d_in[7];
  const float* W1r  = (const float*)d_in[8];
  const float* b1   = (const float*)d_in[9];
  const float* W2l  = (const float*)d_in[10];
  const float* W2r  = (const float*)d_in[11];
  const float* b2   = (const float*)d_in[12];
  int E1 = in_sizes[1];
  int E2 = in_sizes[3];

  // Workspace layout (floats): [agg1][cnt1][agg2][cnt2][h]
  float* agg1 = (float*)d_ws;
  float* cnt1 = agg1 + (long)N1c * DINc;
  float* agg2 = cnt1 + N1c;
  float* cnt2 = agg2 + (long)N2c * Hc;
  float* h    = cnt2 + N2c;

  long zcount = (long)N1c * DINc + N1c + (long)N2c * Hc + N2c;
  zero_kernel<<<2048, 256, 0, stream>>>(agg1, zcount);

  {  // layer-1 scatter-sum: one wave per edge
    long threads = (long)E1 * 32;
    int blocks = (int)((threads + 255) / 256);
    scatter_kernel<<<blocks, 256, 0, stream>>>(x, src1, dst1, agg1, cnt1, E1, DINc);
  }
  {  // layer-1 fused dual-GEMM + bias + relu -> h [40960, 256]
    int waves  = (N1c / 16) * (Hc / 64);  // 16-row tiles x 64-col strips
    int blocks = waves * 32 / 256;
    sage_gemm_wmma<<<blocks, 256, 0, stream>>>(agg1, cnt1, x, W1l, W1r, b1, h,
                                               DINc, 1);
  }
  {  // layer-2 scatter-sum over h
    long threads = (long)E2 * 32;
    int blocks = (int)((threads + 255) / 256);
    scatter_kernel<<<blocks, 256, 0, stream>>>(h, src2, dst2, agg2, cnt2, E2, Hc);
  }
  {  // layer-2 fused dual-GEMM + bias -> out [4096, 256]
    int waves  = (N2c / 16) * (Hc / 64);
    int blocks = waves * 32 / 256;
    sage_gemm_wmma<<<blocks, 256, 0, stream>>>(agg2, cnt2, h, W2l, W2r, b2,
                                               (float*)d_out, Hc, 0);
  }
}